// GATEncoder_6983616824267
// MI455X (gfx1250) — compile-verified
//
#include <hip/hip_runtime.h>
#include <hip/hip_bf16.h>

#define NEG_SLOPE 0.2f

typedef __attribute__((ext_vector_type(2))) float v2f;
typedef __attribute__((ext_vector_type(8))) float v8f;

// ---------------------------------------------------------------------------
// Utility kernels
// ---------------------------------------------------------------------------
__global__ void zero_f32(float* __restrict__ p, int n) {
  int i = blockIdx.x * blockDim.x + threadIdx.x;
  int stride = gridDim.x * blockDim.x;
  for (; i < n; i += stride) p[i] = 0.0f;
}

// out[n, c] = b2[c]  (seed the final atomic accumulation with the bias)
__global__ void out_init(float* __restrict__ out, const float* __restrict__ b,
                         int total, int cmask) {
  int i = blockIdx.x * blockDim.x + threadIdx.x;
  int stride = gridDim.x * blockDim.x;
  for (; i < total; i += stride) out[i] = b[i & cmask];
}

// h = elu(h + b1), in place, c is a power of two
__global__ void bias_elu(float* __restrict__ h, const float* __restrict__ b,
                         int total, int cmask) {
  int i = blockIdx.x * blockDim.x + threadIdx.x;
  int stride = gridDim.x * blockDim.x;
  for (; i < total; i += stride) {
    float v = h[i] + b[i & cmask];
    h[i] = v > 0.0f ? v : (expf(v) - 1.0f);
  }
}

// ---------------------------------------------------------------------------
// Dense GEMM on the matrix pipe: C[M,Nc] = A[M,K] * B[K,Nc], fp32 via
// V_WMMA_F32_16X16X4_F32 (keeps reference-exact fp32 accumulation).
// One wave per 16x16 tile of C. K must be a multiple of 4, M/Nc of 16.
// ---------------------------------------------------------------------------
__global__ void gemm_f32_wmma(const float* __restrict__ A,
                              const float* __restrict__ B,
                              float* __restrict__ C,
                              int M, int Nc, int K) {
  const int lane = threadIdx.x & 31;
  const int wave = threadIdx.x >> 5;
  const int wpb  = blockDim.x >> 5;
  const int m0 = blockIdx.x * 16;
  const int n0 = (blockIdx.y * wpb + wave) * 16;
  if (m0 >= M || n0 >= Nc) return;

  const int half = lane >> 4;   // 0: K pair {k,k+1}, 1: K pair {k+2,k+3}
  const int l15  = lane & 15;

  v8f acc = {};
  const float* ap = A + (size_t)(m0 + l15) * K + 2 * half;        // A row-major
  const float* bp = B + (size_t)(2 * half) * Nc + n0 + l15;       // B row-major

  for (int k = 0; k < K; k += 4) {
    v2f a, b;
    a.x = ap[0];
    a.y = ap[1];
    b.x = bp[0];
    b.y = bp[(size_t)Nc];
    acc = __builtin_amdgcn_wmma_f32_16x16x4_f32(
        /*neg_a=*/false, a, /*neg_b=*/false, b,
        /*c_mod=*/(short)0, acc, /*reuse_a=*/false, /*reuse_b=*/false);
    ap += 4;
    bp += (size_t)4 * Nc;
  }

  // C layout: VGPR r -> row m0 + 8*half + r, col n0 + l15
  float* cp = C + (size_t)(m0 + 8 * half) * Nc + n0 + l15;
#pragma unroll
  for (int r = 0; r < 8; ++r) cp[(size_t)r * Nc] = acc[r];
}

// ---------------------------------------------------------------------------
// Per-node attention scores: as[n,h] = sum_c xh[n,h,c]*att_s[h,c] (same for ad)
// One wave per node, lane = channel (C == 32 == warpSize).
// ---------------------------------------------------------------------------
template <int H, int C>
__global__ void att_scores(const float* __restrict__ xh,
                           const float* __restrict__ att_s,
                           const float* __restrict__ att_d,
                           float* __restrict__ as_, float* __restrict__ ad_,
                           int n_nodes) {
  const int lane = threadIdx.x & 31;
  const int node = blockIdx.x * (blockDim.x >> 5) + (threadIdx.x >> 5);
  if (node >= n_nodes) return;
#pragma unroll
  for (int h = 0; h < H; ++h) {
    float xv = xh[(size_t)node * (H * C) + h * C + lane];
    float s = xv * att_s[h * C + lane];
    float d = xv * att_d[h * C + lane];
#pragma unroll
    for (int off = 16; off > 0; off >>= 1) {
      s += __shfl_xor(s, off, 32);
      d += __shfl_xor(d, off, 32);
    }
    if (lane == 0) {
      as_[(size_t)node * H + h] = s;
      ad_[(size_t)node * H + h] = d;
    }
  }
}

// ---------------------------------------------------------------------------
// Edge helpers (self-loops appended virtually at e >= E)
// ---------------------------------------------------------------------------
__device__ __forceinline__ void edge_sd(const long long* __restrict__ ei,
                                        int e, int E, int& s, int& d) {
  if (e < E) { s = (int)ei[e]; d = (int)ei[(size_t)E + e]; }
  else       { s = e - E; d = e - E; }
}

// order-preserving float <-> uint for atomicMax-based segment max
__device__ __forceinline__ unsigned fenc(float f) {
  unsigned u = __float_as_uint(f);
  return (u & 0x80000000u) ? ~u : (u | 0x80000000u);
}
__device__ __forceinline__ float fdec(unsigned u) {
  return __uint_as_float((u & 0x80000000u) ? (u & 0x7FFFFFFFu) : ~u);
}

__device__ __forceinline__ float lrelu(float v) {
  return v > 0.0f ? v : NEG_SLOPE * v;
}

// pass 1: per-dst segment max of leaky_relu(as[src]+ad[dst])
__global__ void edge_max_k(const long long* __restrict__ ei, int E, int Ep,
                           const float* __restrict__ as_,
                           const float* __restrict__ ad_,
                           unsigned* __restrict__ emax, int H) {
  long long t = (long long)blockIdx.x * blockDim.x + threadIdx.x;
  if (t >= (long long)Ep * H) return;
  int e = (int)(t / H), h = (int)(t - (long long)(t / H) * H);
  int s, d; edge_sd(ei, e, E, s, d);
  float v = lrelu(as_[(size_t)s * H + h] + ad_[(size_t)d * H + h]);
  atomicMax(&emax[(size_t)d * H + h], fenc(v));
}

// pass 2: denominators den[dst,h] = sum exp(e - emax)
__global__ void edge_expsum_k(const long long* __restrict__ ei, int E, int Ep,
                              const float* __restrict__ as_,
                              const float* __restrict__ ad_,
                              const unsigned* __restrict__ emax,
                              float* __restrict__ den, int H) {
  long long t = (long long)blockIdx.x * blockDim.x + threadIdx.x;
  if (t >= (long long)Ep * H) return;
  int e = (int)(t / H), h = (int)(t - (long long)(t / H) * H);
  int s, d; edge_sd(ei, e, E, s, d);
  float v = lrelu(as_[(size_t)s * H + h] + ad_[(size_t)d * H + h]);
  atomicAdd(&den[(size_t)d * H + h], expf(v - fdec(emax[(size_t)d * H + h])));
}

// pass 3: out[dst,h,c] += alpha * feat[src,h,c]; one wave per edge, lane = c
// (C fixed at 32). Gathers/atomics are whole-cacheline coalesced -> L2-bound.
__global__ void edge_agg_k(const long long* __restrict__ ei, int E, int Ep,
                           const float* __restrict__ feat,
                           const float* __restrict__ as_,
                           const float* __restrict__ ad_,
                           const unsigned* __restrict__ emax,
                           const float* __restrict__ den,
                           float* __restrict__ outp, int H) {
  const int lane = threadIdx.x & 31;
  const int e = blockIdx.x * (blockDim.x >> 5) + (threadIdx.x >> 5);
  if (e >= Ep) return;
  int s, d; edge_sd(ei, e, E, s, d);
  for (int h = 0; h < H; ++h) {
    float v = lrelu(as_[(size_t)s * H + h] + ad_[(size_t)d * H + h]);
    float alpha = expf(v - fdec(emax[(size_t)d * H + h])) / den[(size_t)d * H + h];
    atomicAdd(&outp[(size_t)d * (H * 32) + h * 32 + lane],
              feat[(size_t)s * (H * 32) + h * 32 + lane] * alpha);
  }
}

// ---------------------------------------------------------------------------
// Launch
// ---------------------------------------------------------------------------
extern "C" void kernel_launch(void* const* d_in, const int* in_sizes, int n_in,
                              void* d_out, int out_size, void* d_ws, size_t ws_size,
                              hipStream_t stream) {
  (void)n_in; (void)out_size; (void)ws_size;

  const float*     x    = (const float*)d_in[0];
  const long long* ei   = (const long long*)d_in[1];
  const float*     W1   = (const float*)d_in[2];
  const float*     ats1 = (const float*)d_in[3];
  const float*     atd1 = (const float*)d_in[4];
  const float*     b1   = (const float*)d_in[5];
  const float*     W2   = (const float*)d_in[6];
  const float*     ats2 = (const float*)d_in[7];
  const float*     atd2 = (const float*)d_in[8];
  const float*     b2   = (const float*)d_in[9];
  float*           out  = (float*)d_out;

  const int IN = 128, HC = 128, H1 = 4, OUT = 32;
  const int N  = in_sizes[0] / IN;
  const int E  = in_sizes[1] / 2;
  const int Ep = E + N;  // + self loops

  // workspace layout (floats); total = N*308 ~ 123 MB
  float* ws = (float*)d_ws;
  float*    xh    = ws;            ws += (size_t)N * HC;   // layer1 features
  float*    hb    = ws;            ws += (size_t)N * HC;   // agg1 -> elu -> h
  float*    as1   = ws;            ws += (size_t)N * H1;
  float*    ad1   = ws;            ws += (size_t)N * H1;
  unsigned* emax1 = (unsigned*)ws; ws += (size_t)N * H1;
  float*    den1  = ws;            ws += (size_t)N * H1;
  float*    h2    = ws;            ws += (size_t)N * OUT;  // layer2 features
  float*    as2   = ws;            ws += N;
  float*    ad2   = ws;            ws += N;
  unsigned* emax2 = (unsigned*)ws; ws += N;
  float*    den2  = ws;            ws += N;

  auto cdiv = [](long long a, long long b) { return (int)((a + b - 1) / b); };

  // ---- init accumulation buffers ----
  zero_f32<<<cdiv((long long)N * HC, 256), 256, 0, stream>>>(hb, N * HC);
  zero_f32<<<cdiv((long long)N * H1, 256), 256, 0, stream>>>((float*)emax1, N * H1);
  zero_f32<<<cdiv((long long)N * H1, 256), 256, 0, stream>>>(den1, N * H1);
  zero_f32<<<cdiv(N, 256), 256, 0, stream>>>((float*)emax2, N);
  zero_f32<<<cdiv(N, 256), 256, 0, stream>>>(den2, N);
  out_init<<<cdiv((long long)N * OUT, 256), 256, 0, stream>>>(out, b2, N * OUT, OUT - 1);

  // ---- layer 1 ----
  // xh = x @ W1 : grid (6250 row-tiles, 2) x 4 waves -> 8 col tiles of 16
  gemm_f32_wmma<<<dim3(N / 16, 2), 128, 0, stream>>>(x, W1, xh, N, HC, IN);
  att_scores<4, 32><<<cdiv(N, 8), 256, 0, stream>>>(xh, ats1, atd1, as1, ad1, N);
  edge_max_k<<<cdiv((long long)Ep * H1, 256), 256, 0, stream>>>(ei, E, Ep, as1, ad1, emax1, H1);
  edge_expsum_k<<<cdiv((long long)Ep * H1, 256), 256, 0, stream>>>(ei, E, Ep, as1, ad1, emax1, den1, H1);
  edge_agg_k<<<cdiv(Ep, 8), 256, 0, stream>>>(ei, E, Ep, xh, as1, ad1, emax1, den1, hb, H1);
  bias_elu<<<cdiv((long long)N * HC, 256), 256, 0, stream>>>(hb, b1, N * HC, HC - 1);

  // ---- layer 2 (heads=1, ch=32) ----
  // h2 = h @ W2 : grid (6250, 1) x 2 waves -> 2 col tiles of 16
  gemm_f32_wmma<<<dim3(N / 16, 1), 64, 0, stream>>>(hb, W2, h2, N, OUT, HC);
  att_scores<1, 32><<<cdiv(N, 8), 256, 0, stream>>>(h2, ats2, atd2, as2, ad2, N);
  edge_max_k<<<cdiv(Ep, 256), 256, 0, stream>>>(ei, E, Ep, as2, ad2, emax2, 1);
  edge_expsum_k<<<cdiv(Ep, 256), 256, 0, stream>>>(ei, E, Ep, as2, ad2, emax2, den2, 1);
  edge_agg_k<<<cdiv(Ep, 8), 256, 0, stream>>>(ei, E, Ep, h2, as2, ad2, emax2, den2, out, 1);
}